// BinDGCNN_BF1_12635793784944
// MI455X (gfx1250) — compile-verified
//
#include <hip/hip_runtime.h>
#include <math.h>

#define NPTS 1024
#define NB   8
#define KNNK 20

typedef __attribute__((ext_vector_type(2))) float v2f;
typedef __attribute__((ext_vector_type(8))) float v8f;

// ---------------------------------------------------------------------------
// CDNA5 fp32 WMMA: D(16x16) = A(16x4) * B(4x16) + C
// A frag (32-bit A 16x4, ISA 7.12.2): lane l, vgpr r -> M = l&15, K = r + (l<16?0:2)
// B frag (rows striped across lanes): lane l, vgpr r -> N = l&15, K = r + (l<16?0:2)
// C/D:  lane l, vgpr v -> N = l&15, M = v + (l<16?0:8)
// ---------------------------------------------------------------------------
__device__ __forceinline__ v8f wmma_f32x4(v2f a, v2f b, v8f c) {
  return __builtin_amdgcn_wmma_f32_16x16x4_f32(false, a, false, b, (short)0, c,
                                               false, false);
}

// ---------------------------------------------------------------------------
// sign(w) precompute
// ---------------------------------------------------------------------------
__global__ void sign_kernel(const float* __restrict__ w, float* __restrict__ sw, int n) {
  int i = blockIdx.x * 256 + threadIdx.x;
  if (i < n) sw[i] = (w[i] >= 0.f) ? 1.f : -1.f;
}

// ---------------------------------------------------------------------------
// squared norms per point: xx[b,n] = sum_c X[b,c,n]^2
// ---------------------------------------------------------------------------
__global__ void xx_kernel(const float* __restrict__ X, long batchStride,
                          float* __restrict__ xx, int C, int N) {
  int i = blockIdx.x * 256 + threadIdx.x;
  if (i >= NB * N) return;
  int b = i / N, n = i % N;
  const float* p = X + (long)b * batchStride + n;
  float s = 0.f;
  for (int c = 0; c < C; ++c) { float v = p[(long)c * N]; s += v * v; }
  xx[i] = s;
}

// ---------------------------------------------------------------------------
// KNN: 64 query rows per block, 8 waves; Gram tiles via fp32 WMMA, chunked
// neg_dist into LDS, per-row top-20 insertion sort in registers.
// ---------------------------------------------------------------------------
template <int C>
__global__ void knn_kernel(const float* __restrict__ X, long batchStride,
                           const float* __restrict__ xx, int* __restrict__ idx, int N) {
  const int b  = blockIdx.y;
  const int r0 = blockIdx.x * 64;
  const float* Xb  = X + (long)b * batchStride;
  const float* xxb = xx + b * N;
  __shared__ float tile[64 * 64];
  const int tid = threadIdx.x, lane = tid & 31, wave = tid >> 5;
  const int koff = (lane < 16) ? 0 : 2;
  const int lan15 = lane & 15;

  float vals[KNNK]; int ids[KNNK];
#pragma unroll
  for (int j = 0; j < KNNK; ++j) { vals[j] = -3.4e38f; ids[j] = 0; }

  for (int c0 = 0; c0 < N; c0 += 64) {
#pragma unroll
    for (int rep = 0; rep < 2; ++rep) {
      int T  = wave + rep * 8;           // 16 tiles = 4 row-tiles x 4 col-tiles
      int rt = T >> 2, ct = T & 3;
      int m0 = r0 + rt * 16, n0 = c0 + ct * 16;
      v8f acc = {0.f, 0.f, 0.f, 0.f, 0.f, 0.f, 0.f, 0.f};
      for (int k = 0; k < C; k += 4) {
        int k0 = k + koff, k1 = k0 + 1;
        v2f a, bb;
        a[0]  = (k0 < C) ? Xb[(long)k0 * N + m0 + lan15] : 0.f;
        a[1]  = (k1 < C) ? Xb[(long)k1 * N + m0 + lan15] : 0.f;
        bb[0] = (k0 < C) ? Xb[(long)k0 * N + n0 + lan15] : 0.f;
        bb[1] = (k1 < C) ? Xb[(long)k1 * N + n0 + lan15] : 0.f;
        acc = wmma_f32x4(a, bb, acc);
      }
#pragma unroll
      for (int v = 0; v < 8; ++v) {
        int rl = rt * 16 + v + ((lane < 16) ? 0 : 8);
        int cl = ct * 16 + lan15;
        tile[rl * 64 + cl] = 2.f * acc[v] - xxb[r0 + rl] - xxb[c0 + cl];
      }
    }
    __syncthreads();
    if (tid < 64) {
      for (int j = 0; j < 64; ++j) {
        float v = tile[tid * 64 + j];
        if (v > vals[KNNK - 1]) {          // strict > : earlier index wins ties
          vals[KNNK - 1] = v; ids[KNNK - 1] = c0 + j;
#pragma unroll
          for (int q = KNNK - 1; q > 0; --q) {
            if (vals[q] > vals[q - 1]) {
              float tv = vals[q]; vals[q] = vals[q - 1]; vals[q - 1] = tv;
              int   ti = ids[q];  ids[q]  = ids[q - 1];  ids[q - 1]  = ti;
            }
          }
        }
      }
    }
    __syncthreads();
  }
  if (tid < 64) {
    int n = r0 + tid;
#pragma unroll
    for (int j = 0; j < KNNK; ++j) idx[((long)b * N + n) * KNNK + j] = ids[j];
  }
}

// ---------------------------------------------------------------------------
// Edge conv: y[b,o,n] = max_k prelu(s[o,n,k] * sum_c signW[o,c]*f[c,n,k])
// f = [x[idx]-x[n] ; x[n]]  (2*CI channels). 16 points/block, WMMA GEMM.
// ---------------------------------------------------------------------------
template <int CI, int CO>
__global__ void edgeconv_kernel(const float* __restrict__ X, long batchStride,
                                const int* __restrict__ idx,
                                const float* __restrict__ sw,
                                const float* __restrict__ s,
                                const float* __restrict__ aptr,
                                float* __restrict__ ypre, int N) {
  constexpr int TWO = 2 * CI;
  constexpr int KP  = (TWO + 3) & ~3;
  const int b  = blockIdx.y;
  const int n0 = blockIdx.x * 16;
  const float* Xb = X + (long)b * batchStride;
  const float alpha = aptr[0];
  __shared__ float fl[KP * 16];
  __shared__ float ym[CO * 16];
  const int tid = threadIdx.x, lane = tid & 31, wave = tid >> 5;
  const int koff = (lane < 16) ? 0 : 2;
  const int lan15 = lane & 15;

  for (int e = tid; e < CO * 16; e += 256) ym[e] = -3.4e38f;
  __syncthreads();

  for (int t = 0; t < KNNK; ++t) {
    for (int e = tid; e < KP * 16; e += 256) {
      int c = e >> 4, i = e & 15, n = n0 + i;
      float val = 0.f;
      if (c < CI) {
        int j = idx[((long)b * N + n) * KNNK + t];
        val = Xb[(long)c * N + j] - Xb[(long)c * N + n];
      } else if (c < TWO) {
        val = Xb[(long)(c - CI) * N + n];
      }
      fl[c * 16 + i] = val;
    }
    __syncthreads();
    for (int mt = wave; mt * 16 < CO; mt += 8) {
      int m0 = mt * 16;
      int row = m0 + lan15;
      v8f acc = {0.f, 0.f, 0.f, 0.f, 0.f, 0.f, 0.f, 0.f};
      for (int k = 0; k < KP; k += 4) {
        int k0 = k + koff, k1 = k0 + 1;
        v2f a, bb;
        a[0]  = (k0 < TWO) ? sw[row * TWO + k0] : 0.f;
        a[1]  = (k1 < TWO) ? sw[row * TWO + k1] : 0.f;
        bb[0] = fl[k0 * 16 + lan15];
        bb[1] = fl[k1 * 16 + lan15];
        acc = wmma_f32x4(a, bb, acc);
      }
#pragma unroll
      for (int v = 0; v < 8; ++v) {
        int o = m0 + v + ((lane < 16) ? 0 : 8);
        int n = n0 + lan15;
        float z = acc[v] * s[((long)o * N + n) * KNNK + t];
        z = (z >= 0.f) ? z : alpha * z;
        int off = o * 16 + lan15;
        ym[off] = fmaxf(ym[off], z);   // (o,i) owned by one thread each t
      }
    }
    __syncthreads();
  }
  for (int e = tid; e < CO * 16; e += 256) {
    int o = e >> 4, i = e & 15;
    ypre[((long)b * CO + o) * N + n0 + i] = ym[e];
  }
}

// ---------------------------------------------------------------------------
// y5 = prelu(sign(w5)(1024x512) @ xcat(b,512,n) * s5)  via WMMA
// ---------------------------------------------------------------------------
__global__ void gemm5_kernel(const float* __restrict__ xcat,
                             const float* __restrict__ sw,
                             const float* __restrict__ s5,
                             const float* __restrict__ aptr,
                             float* __restrict__ y5, int N) {
  const int b = blockIdx.y, n0 = blockIdx.x * 16;
  const float* Xb = xcat + (long)b * 512 * N;
  const float alpha = aptr[0];
  __shared__ float fl[512 * 16];
  const int tid = threadIdx.x, lane = tid & 31, wave = tid >> 5;
  const int koff = (lane < 16) ? 0 : 2;
  const int lan15 = lane & 15;
  for (int e = tid; e < 512 * 16; e += 256) {
    int c = e >> 4, i = e & 15;
    fl[e] = Xb[(long)c * N + n0 + i];
  }
  __syncthreads();
  for (int mt = wave; mt < 64; mt += 8) {
    int m0 = mt * 16;
    int row = m0 + lan15;
    v8f acc = {0.f, 0.f, 0.f, 0.f, 0.f, 0.f, 0.f, 0.f};
    for (int k = 0; k < 512; k += 4) {
      int k0 = k + koff;
      v2f a, bb;
      a[0]  = sw[row * 512 + k0];
      a[1]  = sw[row * 512 + k0 + 1];
      bb[0] = fl[k0 * 16 + lan15];
      bb[1] = fl[(k0 + 1) * 16 + lan15];
      acc = wmma_f32x4(a, bb, acc);
    }
#pragma unroll
    for (int v = 0; v < 8; ++v) {
      int e = m0 + v + ((lane < 16) ? 0 : 8);
      int n = n0 + lan15;
      float z = acc[v] * s5[(long)e * N + n];
      z = (z >= 0.f) ? z : alpha * z;
      y5[((long)b * 1024 + e) * N + n] = z;
    }
  }
}

// ---------------------------------------------------------------------------
// BN over (B,N): per-channel scale/shift (biased variance, as jnp.var)
// ---------------------------------------------------------------------------
__global__ void bnstats_kernel(const float* __restrict__ y,
                               const float* __restrict__ g, const float* __restrict__ be,
                               float* __restrict__ scale, float* __restrict__ shift,
                               int C, int N) {
  int c = blockIdx.x;
  __shared__ float ssum[256], ssq[256];
  float s = 0.f, q = 0.f;
  for (int b = 0; b < NB; ++b) {
    const float* p = y + ((long)b * C + c) * N;
    for (int n = threadIdx.x; n < N; n += 256) { float v = p[n]; s += v; q += v * v; }
  }
  ssum[threadIdx.x] = s; ssq[threadIdx.x] = q;
  __syncthreads();
  for (int st = 128; st > 0; st >>= 1) {
    if (threadIdx.x < st) {
      ssum[threadIdx.x] += ssum[threadIdx.x + st];
      ssq[threadIdx.x]  += ssq[threadIdx.x + st];
    }
    __syncthreads();
  }
  if (threadIdx.x == 0) {
    float M = (float)(NB * N);
    float mean = ssum[0] / M;
    float var  = ssq[0] / M - mean * mean;
    float sc = g[c] * rsqrtf(var + 1e-5f);
    scale[c] = sc; shift[c] = be[c] - mean * sc;
  }
}

__global__ void bnapply_kernel(const float* __restrict__ y,
                               const float* __restrict__ sc, const float* __restrict__ sh,
                               float* __restrict__ out, int C, int N,
                               int outCtot, int outOff) {
  long i = (long)blockIdx.x * 256 + threadIdx.x;
  long total = (long)NB * C * N;
  if (i >= total) return;
  int n = (int)(i % N); long r = i / N; int c = (int)(r % C); int b = (int)(r / C);
  out[((long)b * outCtot + outOff + c) * N + n] = y[i] * sc[c] + sh[c];
}

// ---------------------------------------------------------------------------
// pooling: h[b,e] = max_n y5n, h[b,1024+e] = mean_n y5n
// ---------------------------------------------------------------------------
__global__ void pool_kernel(const float* __restrict__ y5, float* __restrict__ h, int N) {
  int e = blockIdx.x, b = blockIdx.y;
  const float* p = y5 + ((long)b * 1024 + e) * N;
  __shared__ float sm[256], ss[256];
  float mx = -3.4e38f, sum = 0.f;
  for (int n = threadIdx.x; n < N; n += 256) { float v = p[n]; mx = fmaxf(mx, v); sum += v; }
  sm[threadIdx.x] = mx; ss[threadIdx.x] = sum;
  __syncthreads();
  for (int st = 128; st > 0; st >>= 1) {
    if (threadIdx.x < st) {
      sm[threadIdx.x] = fmaxf(sm[threadIdx.x], sm[threadIdx.x + st]);
      ss[threadIdx.x] += ss[threadIdx.x + st];
    }
    __syncthreads();
  }
  if (threadIdx.x == 0) { h[b * 2048 + e] = sm[0]; h[b * 2048 + 1024 + e] = ss[0] / (float)N; }
}

// ---------------------------------------------------------------------------
// tiny MLP tail
// ---------------------------------------------------------------------------
__global__ void lingemm_kernel(const float* __restrict__ h, const float* __restrict__ sw,
                               float* __restrict__ out, int CIn, int COut) {
  int i = blockIdx.x * 256 + threadIdx.x;
  if (i >= NB * COut) return;
  int b = i / COut, o = i % COut;
  const float* hp = h + (long)b * CIn;
  const float* wp = sw + (long)o * CIn;
  float s = 0.f;
  for (int c = 0; c < CIn; ++c) s += hp[c] * wp[c];
  out[i] = s;
}

__global__ void bnc_kernel(float* __restrict__ x, const float* __restrict__ g,
                           const float* __restrict__ be, const float* __restrict__ aptr,
                           int CO, int doprelu) {
  int o = blockIdx.x * 256 + threadIdx.x;
  if (o >= CO) return;
  float m = 0.f, q = 0.f;
  for (int b = 0; b < NB; ++b) { float v = x[b * CO + o]; m += v; q += v * v; }
  m /= (float)NB;
  float var = q / (float)NB - m * m;
  float sc = g[o] * rsqrtf(var + 1e-5f);
  float sh = be[o] - m * sc;
  float alpha = doprelu ? aptr[0] : 0.f;
  for (int b = 0; b < NB; ++b) {
    float v = x[b * CO + o] * sc + sh;
    if (doprelu) v = (v >= 0.f) ? v : alpha * v;
    x[b * CO + o] = v;
  }
}

// ---------------------------------------------------------------------------
extern "C" void kernel_launch(void* const* d_in, const int* in_sizes, int n_in,
                              void* d_out, int out_size, void* d_ws, size_t ws_size,
                              hipStream_t stream) {
  (void)in_sizes; (void)n_in; (void)out_size; (void)ws_size;
  const int N = NPTS;
  float* ws = (float*)d_ws;
  size_t off = 0;
  auto alloc = [&](size_t nElem) { float* p = ws + off; off += nElem; return p; };

  float* xcat  = alloc((size_t)NB * 512 * N);     // concat x1..x4, (B,512,N)
  float* y5    = alloc((size_t)NB * 1024 * N);
  float* ypre  = alloc((size_t)NB * 256 * N);
  int*   idxb  = (int*)alloc((size_t)NB * N * KNNK);
  float* xx    = alloc((size_t)NB * N);
  float* scale = alloc(1024);
  float* shift = alloc(1024);
  float* sw1   = alloc(64 * 6);
  float* sw2   = alloc(64 * 128);
  float* sw3   = alloc(128 * 128);
  float* sw4   = alloc(256 * 256);
  float* sw5   = alloc(1024 * 512);
  float* slw1  = alloc(512 * 2048);
  float* slw2  = alloc(256 * 512);
  float* slw3  = alloc(40 * 256);
  float* h     = alloc(NB * 2048);
  float* h2    = alloc(NB * 512);
  float* h3    = alloc(NB * 256);

  auto F = [&](int i) { return (const float*)d_in[i]; };
  auto sign = [&](const float* w, float* sw, int n) {
    sign_kernel<<<(n + 255) / 256, 256, 0, stream>>>(w, sw, n);
  };
  sign(F(1),  sw1,  64 * 6);
  sign(F(6),  sw2,  64 * 128);
  sign(F(11), sw3,  128 * 128);
  sign(F(16), sw4,  256 * 256);
  sign(F(21), sw5,  1024 * 512);
  sign(F(26), slw1, 512 * 2048);
  sign(F(30), slw2, 256 * 512);
  sign(F(34), slw3, 40 * 256);

  const dim3 gKnn(N / 64, NB), gCv(N / 16, NB);
  const int xxg = (NB * N + 255) / 256;

  // ---- block 1: x0 (B,3,N) -> 64 ch at xcat[:,0:64,:]
  {
    const float* X = F(0); long bs = 3l * N;
    xx_kernel<<<xxg, 256, 0, stream>>>(X, bs, xx, 3, N);
    knn_kernel<3><<<gKnn, 256, 0, stream>>>(X, bs, xx, idxb, N);
    edgeconv_kernel<3, 64><<<gCv, 256, 0, stream>>>(X, bs, idxb, sw1, F(2), F(3), ypre, N);
    bnstats_kernel<<<64, 256, 0, stream>>>(ypre, F(4), F(5), scale, shift, 64, N);
    bnapply_kernel<<<(NB * 64 * N + 255) / 256, 256, 0, stream>>>(ypre, scale, shift, xcat, 64, N, 512, 0);
  }
  // ---- block 2: x1 (64 ch @ off 0) -> 64 ch at off 64
  {
    const float* X = xcat; long bs = 512l * N;
    xx_kernel<<<xxg, 256, 0, stream>>>(X, bs, xx, 64, N);
    knn_kernel<64><<<gKnn, 256, 0, stream>>>(X, bs, xx, idxb, N);
    edgeconv_kernel<64, 64><<<gCv, 256, 0, stream>>>(X, bs, idxb, sw2, F(7), F(8), ypre, N);
    bnstats_kernel<<<64, 256, 0, stream>>>(ypre, F(9), F(10), scale, shift, 64, N);
    bnapply_kernel<<<(NB * 64 * N + 255) / 256, 256, 0, stream>>>(ypre, scale, shift, xcat, 64, N, 512, 64);
  }
  // ---- block 3: x2 (64 ch @ off 64) -> 128 ch at off 128
  {
    const float* X = xcat + 64l * N; long bs = 512l * N;
    xx_kernel<<<xxg, 256, 0, stream>>>(X, bs, xx, 64, N);
    knn_kernel<64><<<gKnn, 256, 0, stream>>>(X, bs, xx, idxb, N);
    edgeconv_kernel<64, 128><<<gCv, 256, 0, stream>>>(X, bs, idxb, sw3, F(12), F(13), ypre, N);
    bnstats_kernel<<<128, 256, 0, stream>>>(ypre, F(14), F(15), scale, shift, 128, N);
    bnapply_kernel<<<(NB * 128 * N + 255) / 256, 256, 0, stream>>>(ypre, scale, shift, xcat, 128, N, 512, 128);
  }
  // ---- block 4: x3 (128 ch @ off 128) -> 256 ch at off 256
  {
    const float* X = xcat + 128l * N; long bs = 512l * N;
    xx_kernel<<<xxg, 256, 0, stream>>>(X, bs, xx, 128, N);
    knn_kernel<128><<<gKnn, 256, 0, stream>>>(X, bs, xx, idxb, N);
    edgeconv_kernel<128, 256><<<gCv, 256, 0, stream>>>(X, bs, idxb, sw4, F(17), F(18), ypre, N);
    bnstats_kernel<<<256, 256, 0, stream>>>(ypre, F(19), F(20), scale, shift, 256, N);
    bnapply_kernel<<<(NB * 256 * N + 255) / 256, 256, 0, stream>>>(ypre, scale, shift, xcat, 256, N, 512, 256);
  }
  // ---- global binarized projection + BN + pooling
  gemm5_kernel<<<gCv, 256, 0, stream>>>(xcat, sw5, F(22), F(23), y5, N);
  bnstats_kernel<<<1024, 256, 0, stream>>>(y5, F(24), F(25), scale, shift, 1024, N);
  bnapply_kernel<<<(int)(((long)NB * 1024 * N + 255) / 256), 256, 0, stream>>>(y5, scale, shift, y5, 1024, N, 1024, 0);
  pool_kernel<<<dim3(1024, NB), 256, 0, stream>>>(y5, h, N);

  // ---- MLP tail
  lingemm_kernel<<<(NB * 512 + 255) / 256, 256, 0, stream>>>(h, slw1, h2, 2048, 512);
  bnc_kernel<<<2, 256, 0, stream>>>(h2, F(27), F(28), F(29), 512, 1);
  lingemm_kernel<<<(NB * 256 + 255) / 256, 256, 0, stream>>>(h2, slw2, h3, 512, 256);
  bnc_kernel<<<1, 256, 0, stream>>>(h3, F(31), F(32), F(33), 256, 1);
  lingemm_kernel<<<(NB * 40 + 255) / 256, 256, 0, stream>>>(h3, slw3, (float*)d_out, 256, 40);
  bnc_kernel<<<1, 256, 0, stream>>>((float*)d_out, F(35), F(36), nullptr, 40, 0);
}